// EdgeLayer_43430709297206
// MI455X (gfx1250) — compile-verified
//
#include <hip/hip_runtime.h>

typedef __attribute__((ext_vector_type(2))) float v2f;
typedef __attribute__((ext_vector_type(8))) float v8f;

#define EPSN 1e-8f

__device__ __forceinline__ float sigmoidf(float x) { return 1.f / (1.f + __expf(-x)); }

__device__ __forceinline__ v8f vzero8() {
  v8f z = {0.f, 0.f, 0.f, 0.f, 0.f, 0.f, 0.f, 0.f};
  return z;
}

// D(16x16) = A(16x4) * B(4x16) + C, full fp32 WMMA (CDNA5)
__device__ __forceinline__ v8f wmma4(v2f a, v2f b, v8f c) {
  return __builtin_amdgcn_wmma_f32_16x16x4_f32(false, a, false, b, (short)0, c, false, false);
}

// ---------------------------------------------------------------------------
// Prep: repack weight [K,N] into zero-padded K-pair-interleaved [KP/2][N][2]
// so a WMMA lane's two B values are one contiguous 8-byte load, guard-free.
// ---------------------------------------------------------------------------
__global__ void pad_pair_kernel(const float* __restrict__ src, float* __restrict__ dst,
                                int K, int N, int KP2) {
  int i = blockIdx.x * blockDim.x + threadIdx.x;
  int total = KP2 * N * 2;
  if (i >= total) return;
  int j  = i & 1;
  int n  = (i >> 1) % N;
  int kp = (i >> 1) / N;
  int k  = kp * 2 + j;
  dst[i] = (k < K) ? src[k * N + n] : 0.f;
}

// ---------------------------------------------------------------------------
// Kernel 1: gather + 3 msg GVP layers -> smsg [E,32], vmsg [E,3] in workspace
// One wave32 per 16-edge tile.
// ---------------------------------------------------------------------------
__global__ __launch_bounds__(32)
void gvp_msg_kernel(const float* __restrict__ hVs, const float* __restrict__ hVv,
                    const float* __restrict__ hEs, const float* __restrict__ hEv,
                    const int* __restrict__ eidx,
                    const float* __restrict__ wh0, const float* __restrict__ pws0,
                    const float* __restrict__ bs0, const float* __restrict__ wv0,
                    const float* __restrict__ wh1, const float* __restrict__ pws1,
                    const float* __restrict__ bs1, const float* __restrict__ wv1,
                    const float* __restrict__ wh2, const float* __restrict__ pws2,
                    const float* __restrict__ bs2, const float* __restrict__ wv2,
                    float* __restrict__ smsg, float* __restrict__ vmsg, int E)
{
  __shared__ float featA[16 * 268];  // [s(232) | vn(33) | pad(3)] per edge, K-padded
  __shared__ float vcat[16 * 99];    // concat vectors [e][33][3]
  __shared__ float featB[16 * 36];   // [s(32) | vn(1) | pad(3)] for layers 1/2
  __shared__ float pvo[2 * 16 * 3];  // partial vo reduction
  __shared__ float vn2s[16];
  __shared__ int   sd[32];           // src[16], dst[16]
  __shared__ float outS[16 * 32];

  const int lane = threadIdx.x;
  const int e0 = blockIdx.x * 16;

  if (lane < 16) {
    int e = e0 + lane; if (e >= E) e = E - 1;
    sd[lane]      = eidx[e];
    sd[16 + lane] = eidx[E + e];
  }
  __syncthreads();

  // stage scalar features [hVs[src](100) | hEs(32) | hVs[dst](100)]
  for (int idx = lane; idx < 16 * 232; idx += 32) {
    int el = idx / 232, c = idx - el * 232;
    int e = e0 + el; if (e >= E) e = E - 1;
    float v;
    if (c < 100)      v = hVs[sd[el] * 100 + c];
    else if (c < 132) v = hEs[e * 32 + (c - 100)];
    else              v = hVs[sd[16 + el] * 100 + (c - 132)];
    featA[el * 268 + c] = v;
  }
  for (int idx = lane; idx < 16 * 3; idx += 32) {   // zero K pad 265..267
    int el = idx / 3;
    featA[el * 268 + 265 + (idx - el * 3)] = 0.f;
  }
  // stage concat vectors [hVv[src](16) | hEv(1) | hVv[dst](16)] x 3
  for (int idx = lane; idx < 16 * 99; idx += 32) {
    int el = idx / 99, r = idx - el * 99;
    int vch = r / 3, d = r - vch * 3;
    int e = e0 + el; if (e >= E) e = E - 1;
    float val;
    if (vch < 16)       val = hVv[(sd[el] * 16 + vch) * 3 + d];
    else if (vch == 16) val = hEv[e * 3 + d];
    else                val = hVv[(sd[16 + el] * 16 + (vch - 17)) * 3 + d];
    vcat[el * 99 + r] = val;
  }
  __syncthreads();

  // vh = vcat @ wh0 : per-lane subset of h channels; vn -> featA cols 232..264
  {
    const int el = lane & 15, rp = lane >> 4;
    const int hb = rp ? 17 : 0, he = rp ? 33 : 17;
    float vox = 0.f, voy = 0.f, voz = 0.f;
    for (int h = hb; h < he; ++h) {
      float x = 0.f, y = 0.f, z = 0.f;
      for (int v = 0; v < 33; ++v) {
        float w = wh0[v * 33 + h];
        const float* p = &vcat[el * 99 + v * 3];
        x = fmaf(p[0], w, x); y = fmaf(p[1], w, y); z = fmaf(p[2], w, z);
      }
      featA[el * 268 + 232 + h] = sqrtf(fmaxf(x * x + y * y + z * z, EPSN));
      float w = wv0[h];
      vox = fmaf(x, w, vox); voy = fmaf(y, w, voy); voz = fmaf(z, w, voz);
    }
    pvo[(rp * 16 + el) * 3 + 0] = vox;
    pvo[(rp * 16 + el) * 3 + 1] = voy;
    pvo[(rp * 16 + el) * 3 + 2] = voz;
  }
  __syncthreads();

  // small per-edge vector chain through all three msg layers (VO = 1)
  if (lane < 16) {
    float x = pvo[lane * 3 + 0] + pvo[(16 + lane) * 3 + 0];
    float y = pvo[lane * 3 + 1] + pvo[(16 + lane) * 3 + 1];
    float z = pvo[lane * 3 + 2] + pvo[(16 + lane) * 3 + 2];
    float g = sigmoidf(sqrtf(fmaxf(x * x + y * y + z * z, EPSN)));  // msg0 act
    x *= g; y *= g; z *= g;
    float w = wh1[0];                                               // msg1
    float ax = x * w, ay = y * w, az = z * w;
    featB[lane * 36 + 32] = sqrtf(fmaxf(ax * ax + ay * ay + az * az, EPSN));
    featB[lane * 36 + 33] = 0.f; featB[lane * 36 + 34] = 0.f; featB[lane * 36 + 35] = 0.f;
    w = wv1[0];
    ax *= w; ay *= w; az *= w;
    g = sigmoidf(sqrtf(fmaxf(ax * ax + ay * ay + az * az, EPSN)));
    ax *= g; ay *= g; az *= g;
    w = wh2[0];                                                     // msg2
    float bx = ax * w, by = ay * w, bz = az * w;
    vn2s[lane] = sqrtf(fmaxf(bx * bx + by * by + bz * bz, EPSN));
    w = wv2[0];
    int e = e0 + lane;
    if (e < E) {
      vmsg[e * 3 + 0] = bx * w; vmsg[e * 3 + 1] = by * w; vmsg[e * 3 + 2] = bz * w;
    }
  }
  __syncthreads();

  const int m    = lane & 15;        // edge row within tile (A) / output col (B,D)
  const int rp   = lane >> 4;
  const int koff = rp * 2;           // K sub-offset per half-wave
  const int nn   = lane & 15;
  const int mofs = rp * 8;           // D row offset

  // per-lane base offsets into pair-interleaved weights [kpair][N][2]
  const int bofs32 = (rp * 32 + nn) * 2;   // N=32 layouts; +32 floats for n-tile 1

  // ---- msg layer 0: s1 = relu([s|vn](16x268) @ ws0(265x32) + bs0) ----
  v8f acc0 = vzero8(), acc1 = vzero8();
  #pragma unroll 4
  for (int t = 0; t < 67; ++t) {
    v2f a  = *(const v2f*)&featA[m * 268 + koff + t * 4];
    v2f b0 = *(const v2f*)(pws0 + bofs32 + t * 128);
    v2f b1 = *(const v2f*)(pws0 + bofs32 + t * 128 + 32);
    acc0 = wmma4(a, b0, acc0);
    acc1 = wmma4(a, b1, acc1);
  }
  for (int r = 0; r < 8; ++r) {
    featB[(r + mofs) * 36 + nn]      = fmaxf(acc0[r] + bs0[nn], 0.f);
    featB[(r + mofs) * 36 + 16 + nn] = fmaxf(acc1[r] + bs0[16 + nn], 0.f);
  }
  __syncthreads();

  // ---- msg layer 1: s2 = relu([s1|vn1](16x36) @ ws1(33x32) + bs1) ----
  acc0 = vzero8(); acc1 = vzero8();
  #pragma unroll 3
  for (int t = 0; t < 9; ++t) {
    v2f a  = *(const v2f*)&featB[m * 36 + koff + t * 4];
    v2f b0 = *(const v2f*)(pws1 + bofs32 + t * 128);
    v2f b1 = *(const v2f*)(pws1 + bofs32 + t * 128 + 32);
    acc0 = wmma4(a, b0, acc0);
    acc1 = wmma4(a, b1, acc1);
  }
  __syncthreads();
  for (int r = 0; r < 8; ++r) {
    featB[(r + mofs) * 36 + nn]      = fmaxf(acc0[r] + bs1[nn], 0.f);
    featB[(r + mofs) * 36 + 16 + nn] = fmaxf(acc1[r] + bs1[16 + nn], 0.f);
  }
  if (lane < 16) featB[lane * 36 + 32] = vn2s[lane];
  __syncthreads();

  // ---- msg layer 2: s3 = [s2|vn2](16x36) @ ws2(33x32) + bs2  (no act) ----
  acc0 = vzero8(); acc1 = vzero8();
  #pragma unroll 3
  for (int t = 0; t < 9; ++t) {
    v2f a  = *(const v2f*)&featB[m * 36 + koff + t * 4];
    v2f b0 = *(const v2f*)(pws2 + bofs32 + t * 128);
    v2f b1 = *(const v2f*)(pws2 + bofs32 + t * 128 + 32);
    acc0 = wmma4(a, b0, acc0);
    acc1 = wmma4(a, b1, acc1);
  }
  for (int r = 0; r < 8; ++r) {
    outS[(r + mofs) * 32 + nn]      = acc0[r] + bs2[nn];
    outS[(r + mofs) * 32 + 16 + nn] = acc1[r] + bs2[16 + nn];
  }
  __syncthreads();
  for (int idx = lane; idx < 512; idx += 32) {   // coalesced store
    int el = idx >> 5;
    int e = e0 + el;
    if (e < E) smsg[e * 32 + (idx & 31)] = outS[idx];
  }
}

// ---------------------------------------------------------------------------
// Kernel 2: merge + residual + LN, 2 ff GVP layers, residual + LN -> outputs
// ---------------------------------------------------------------------------
__global__ __launch_bounds__(32)
void gvp_ff_kernel(const float* __restrict__ hEs, const float* __restrict__ hEv,
                   const float* __restrict__ smsg, const float* __restrict__ vmsg,
                   const float* __restrict__ whf0, const float* __restrict__ pwsf0,
                   const float* __restrict__ bsf0, const float* __restrict__ wvf0,
                   const float* __restrict__ whf1, const float* __restrict__ pwsf1,
                   const float* __restrict__ bsf1, const float* __restrict__ wvf1,
                   const float* __restrict__ g0, const float* __restrict__ b0,
                   const float* __restrict__ g1, const float* __restrict__ b1,
                   float* __restrict__ out, int E)
{
  __shared__ float xsA[16 * 36];     // [xs(32) | vn_ff0(2) | pad(2)]
  __shared__ float featC[16 * 132];  // [ff0 out(128) | vn_ff1(2) | pad(2)]
  __shared__ float xsKeep[16 * 32];  // post-LN1 xs (ff residual source)
  __shared__ float pre2[16 * 32];    // pre-LN2
  __shared__ float mus[16], rstds[16];

  const int lane = threadIdx.x;
  const int e0 = blockIdx.x * 16;
  const int half = E >> 1;

  // merge duplicate edges (reverse of e is (e + E/2) % E) + residual
  for (int idx = lane; idx < 512; idx += 32) {
    int el = idx >> 5, c = idx & 31;
    int e = e0 + el; if (e >= E) e = E - 1;
    int er = e + half; if (er >= E) er -= E;
    xsKeep[idx] = 0.5f * (smsg[e * 32 + c] + smsg[er * 32 + c]) + hEs[e * 32 + c];
  }
  __syncthreads();

  if (lane < 16) {
    int e = e0 + lane; if (e >= E) e = E - 1;
    // LN1 scalar
    float mu = 0.f;
    for (int c = 0; c < 32; ++c) mu += xsKeep[lane * 32 + c];
    mu *= (1.f / 32.f);
    float var = 0.f;
    for (int c = 0; c < 32; ++c) { float d = xsKeep[lane * 32 + c] - mu; var += d * d; }
    var *= (1.f / 32.f);
    float rstd = rsqrtf(var + 1e-5f);
    for (int c = 0; c < 32; ++c) {
      float xn = (xsKeep[lane * 32 + c] - mu) * rstd * g0[c] + b0[c];
      xsKeep[lane * 32 + c] = xn;
      xsA[lane * 36 + c] = xn;
    }
    // LN1 vector (VO = 1 -> normalize to unit length with EPS clamp)
    float vx = vmsg[e * 3 + 0] + hEv[e * 3 + 0];
    float vy = vmsg[e * 3 + 1] + hEv[e * 3 + 1];
    float vz = vmsg[e * 3 + 2] + hEv[e * 3 + 2];
    float inv = rsqrtf(fmaxf(vx * vx + vy * vy + vz * vz, EPSN));
    vx *= inv; vy *= inv; vz *= inv;
    // ff0 vector branch (h = 2)
    float vhh[2][3];
    for (int h = 0; h < 2; ++h) {
      float w = whf0[h];
      vhh[h][0] = vx * w; vhh[h][1] = vy * w; vhh[h][2] = vz * w;
      xsA[lane * 36 + 32 + h] =
          sqrtf(fmaxf(vhh[h][0] * vhh[h][0] + vhh[h][1] * vhh[h][1] + vhh[h][2] * vhh[h][2], EPSN));
    }
    xsA[lane * 36 + 34] = 0.f; xsA[lane * 36 + 35] = 0.f;
    float vo[2][3];
    for (int o = 0; o < 2; ++o) {
      for (int d = 0; d < 3; ++d)
        vo[o][d] = vhh[0][d] * wvf0[0 * 2 + o] + vhh[1][d] * wvf0[1 * 2 + o];
      float g = sigmoidf(sqrtf(fmaxf(vo[o][0] * vo[o][0] + vo[o][1] * vo[o][1] + vo[o][2] * vo[o][2], EPSN)));
      vo[o][0] *= g; vo[o][1] *= g; vo[o][2] *= g;
    }
    // ff1 vector branch
    float vh1[2][3];
    for (int h = 0; h < 2; ++h) {
      for (int d = 0; d < 3; ++d)
        vh1[h][d] = vo[0][d] * whf1[0 * 2 + h] + vo[1][d] * whf1[1 * 2 + h];
      featC[lane * 132 + 128 + h] =
          sqrtf(fmaxf(vh1[h][0] * vh1[h][0] + vh1[h][1] * vh1[h][1] + vh1[h][2] * vh1[h][2], EPSN));
    }
    featC[lane * 132 + 130] = 0.f; featC[lane * 132 + 131] = 0.f;
    float dvx = vh1[0][0] * wvf1[0] + vh1[1][0] * wvf1[1];
    float dvy = vh1[0][1] * wvf1[0] + vh1[1][1] * wvf1[1];
    float dvz = vh1[0][2] * wvf1[0] + vh1[1][2] * wvf1[1];
    // LN2 vector + output (second half of d_out)
    float ox = vx + dvx, oy = vy + dvy, oz = vz + dvz;
    inv = rsqrtf(fmaxf(ox * ox + oy * oy + oz * oz, EPSN));
    if (e0 + lane < E) {
      out[(size_t)E * 32 + (size_t)e * 3 + 0] = ox * inv;
      out[(size_t)E * 32 + (size_t)e * 3 + 1] = oy * inv;
      out[(size_t)E * 32 + (size_t)e * 3 + 2] = oz * inv;
    }
  }
  __syncthreads();

  const int m    = lane & 15;
  const int rp   = lane >> 4;
  const int koff = rp * 2;
  const int nn   = lane & 15;
  const int mofs = rp * 8;
  const int bofs32  = (rp * 32 + nn) * 2;    // N=32  pair layout base
  const int bofs128 = (rp * 128 + nn) * 2;   // N=128 pair layout base (+32 per n-tile)

  // ---- ff0: relu([xs|vn](16x36) @ wsf0(34x128) + bsf0), N = 128 (8 tiles) ----
  v8f acc[8];
  for (int nt = 0; nt < 8; ++nt) acc[nt] = vzero8();
  #pragma unroll 3
  for (int t = 0; t < 9; ++t) {
    v2f a = *(const v2f*)&xsA[m * 36 + koff + t * 4];
    #pragma unroll
    for (int nt = 0; nt < 8; ++nt) {
      v2f b = *(const v2f*)(pwsf0 + bofs128 + nt * 32 + t * 512);
      acc[nt] = wmma4(a, b, acc[nt]);
    }
  }
  for (int nt = 0; nt < 8; ++nt)
    for (int r = 0; r < 8; ++r)
      featC[(r + mofs) * 132 + nt * 16 + nn] = fmaxf(acc[nt][r] + bsf0[nt * 16 + nn], 0.f);
  __syncthreads();

  // ---- ff1: ds = [ff0|vn](16x132) @ wsf1(130x32) + bsf1  (no act) ----
  v8f d0 = vzero8(), d1 = vzero8();
  #pragma unroll 4
  for (int t = 0; t < 33; ++t) {
    v2f a  = *(const v2f*)&featC[m * 132 + koff + t * 4];
    v2f b0 = *(const v2f*)(pwsf1 + bofs32 + t * 128);
    v2f b1 = *(const v2f*)(pwsf1 + bofs32 + t * 128 + 32);
    d0 = wmma4(a, b0, d0);
    d1 = wmma4(a, b1, d1);
  }
  for (int r = 0; r < 8; ++r) {
    pre2[(r + mofs) * 32 + nn]      = xsKeep[(r + mofs) * 32 + nn]      + d0[r] + bsf1[nn];
    pre2[(r + mofs) * 32 + 16 + nn] = xsKeep[(r + mofs) * 32 + 16 + nn] + d1[r] + bsf1[16 + nn];
  }
  __syncthreads();

  // LN2 scalar
  if (lane < 16) {
    float mu = 0.f;
    for (int c = 0; c < 32; ++c) mu += pre2[lane * 32 + c];
    mu *= (1.f / 32.f);
    float var = 0.f;
    for (int c = 0; c < 32; ++c) { float d = pre2[lane * 32 + c] - mu; var += d * d; }
    var *= (1.f / 32.f);
    mus[lane] = mu; rstds[lane] = rsqrtf(var + 1e-5f);
  }
  __syncthreads();
  for (int idx = lane; idx < 512; idx += 32) {
    int el = idx >> 5, c = idx & 31;
    int e = e0 + el;
    if (e < E) out[(size_t)e * 32 + c] = (pre2[idx] - mus[el]) * rstds[el] * g1[c] + b1[c];
  }
}

// ---------------------------------------------------------------------------
extern "C" void kernel_launch(void* const* d_in, const int* in_sizes, int n_in,
                              void* d_out, int out_size, void* d_ws, size_t ws_size,
                              hipStream_t stream) {
  const float* hVs  = (const float*)d_in[0];
  const float* hVv  = (const float*)d_in[1];
  const float* hEs  = (const float*)d_in[2];
  const float* hEv  = (const float*)d_in[3];
  const int*   eidx = (const int*)d_in[4];
  // params['msg'][0..2]{wh,ws,bs,wv}
  const float* wh0 = (const float*)d_in[5];
  const float* ws0 = (const float*)d_in[6];
  const float* bs0 = (const float*)d_in[7];
  const float* wv0 = (const float*)d_in[8];
  const float* wh1 = (const float*)d_in[9];
  const float* ws1 = (const float*)d_in[10];
  const float* bs1 = (const float*)d_in[11];
  const float* wv1 = (const float*)d_in[12];
  const float* wh2 = (const float*)d_in[13];
  const float* ws2 = (const float*)d_in[14];
  const float* bs2 = (const float*)d_in[15];
  const float* wv2 = (const float*)d_in[16];
  // params['ff'][0..1]{wh,ws,bs,wv}
  const float* whf0 = (const float*)d_in[17];
  const float* wsf0 = (const float*)d_in[18];
  const float* bsf0 = (const float*)d_in[19];
  const float* wvf0 = (const float*)d_in[20];
  const float* whf1 = (const float*)d_in[21];
  const float* wsf1 = (const float*)d_in[22];
  const float* bsf1 = (const float*)d_in[23];
  const float* wvf1 = (const float*)d_in[24];
  // params['ln'][0..1]{g,b}
  const float* g0 = (const float*)d_in[25];
  const float* b0 = (const float*)d_in[26];
  const float* g1 = (const float*)d_in[27];
  const float* b1 = (const float*)d_in[28];

  const int E = in_sizes[4] / 2;

  // workspace layout
  float* smsg = (float*)d_ws;
  float* vmsg = smsg + (size_t)E * 32;
  float* pw   = vmsg + (size_t)E * 3;
  float* pws0  = pw;                  // 134*32*2 = 8576
  float* pws1  = pws0  + 134 * 64;    // 18*32*2  = 1152
  float* pws2  = pws1  + 18 * 64;     // 1152
  float* pwsf0 = pws2  + 18 * 64;     // 18*128*2 = 4608
  float* pwsf1 = pwsf0 + 18 * 256;    // 66*32*2  = 4224

  // repack weights (zero-padded, K-pair interleaved)
  pad_pair_kernel<<<(134 * 64 + 255) / 256, 256, 0, stream>>>(ws0,  pws0,  265, 32,  134);
  pad_pair_kernel<<<(18 * 64 + 255) / 256, 256, 0, stream>>>(ws1,  pws1,  33,  32,  18);
  pad_pair_kernel<<<(18 * 64 + 255) / 256, 256, 0, stream>>>(ws2,  pws2,  33,  32,  18);
  pad_pair_kernel<<<(18 * 256 + 255) / 256, 256, 0, stream>>>(wsf0, pwsf0, 34,  128, 18);
  pad_pair_kernel<<<(66 * 64 + 255) / 256, 256, 0, stream>>>(wsf1, pwsf1, 130, 32,  66);

  const int nTiles = (E + 15) / 16;
  gvp_msg_kernel<<<nTiles, 32, 0, stream>>>(hVs, hVv, hEs, hEv, eidx,
      wh0, pws0, bs0, wv0, wh1, pws1, bs1, wv1, wh2, pws2, bs2, wv2,
      smsg, vmsg, E);
  gvp_ff_kernel<<<nTiles, 32, 0, stream>>>(hEs, hEv, smsg, vmsg,
      whf0, pwsf0, bsf0, wvf0, whf1, pwsf1, bsf1, wvf1,
      g0, b0, g1, b1, (float*)d_out, E);
}